// OrientedRepPointsLoss_58402965291304
// MI455X (gfx1250) — compile-verified
//
#include <hip/hip_runtime.h>
#include <stdint.h>

// CDNA5 / gfx1250: wave32, WMMA f32 16x16x4 for the pairwise-distance GEMM.
typedef __attribute__((ext_vector_type(2))) float v2f;
typedef __attribute__((ext_vector_type(8))) float v8f;

#define KGT 256
#define FULLKEY 0xFFFFFFFFFFFFFFFFull
#define INVBITS 0xFFFFFFFFu

struct Ws {
    int lvlmin;
    int lvlmax;
    int pad0[2];
    unsigned long long gkeys[KGT];   // packed (d2_bits<<32)|point_idx per gt
    float c0[KGT], c1[KGT], c2[KGT], c3[KGT], c4[KGT];
    int   glvl[KGT];
};

__device__ __forceinline__ int imin(int a, int b) { return a < b ? a : b; }
__device__ __forceinline__ int imax(int a, int b) { return a > b ? a : b; }

// ---------------- init: zero outputs, reset workspace ----------------
__global__ void k_init(float* __restrict__ out, int twoN, Ws* __restrict__ ws) {
    int i = blockIdx.x * blockDim.x + threadIdx.x;
    if (i < twoN) out[i] = 0.0f;
    if (i < KGT)  ws->gkeys[i] = FULLKEY;
    if (i == 0) { ws->lvlmin = 0x7FFFFFFF; ws->lvlmax = (int)0x80000000; }
}

// ---------------- level min/max over points ----------------
__global__ void k_lvl(const float* __restrict__ pts, int N, Ws* __restrict__ ws) {
    __shared__ int smin[256], smax[256];
    int tid = threadIdx.x;
    int lmin = 0x7FFFFFFF, lmax = (int)0x80000000;
    for (int i = blockIdx.x * blockDim.x + tid; i < N; i += gridDim.x * blockDim.x) {
        float s = pts[3 * i + 2];                       // stride, exact power of 2
        int pl = (int)((__float_as_uint(s) >> 23) & 0xFF) - 127;  // exact log2
        lmin = imin(lmin, pl); lmax = imax(lmax, pl);
    }
    smin[tid] = lmin; smax[tid] = lmax;
    __syncthreads();
    for (int off = 128; off > 0; off >>= 1) {
        if (tid < off) {
            smin[tid] = imin(smin[tid], smin[tid + off]);
            smax[tid] = imax(smax[tid], smax[tid + off]);
        }
        __syncthreads();
    }
    if (tid == 0) { atomicMin(&ws->lvlmin, smin[0]); atomicMax(&ws->lvlmax, smax[0]); }
}

// ---------------- per-gt coefficients + level ----------------
__global__ void k_prep(const float* __restrict__ gtb, Ws* __restrict__ ws) {
    int k = threadIdx.x;                                  // one block of 256
    const float* r = gtb + 8 * k;
    float x0 = r[0], y0 = r[1], x1 = r[2], y1 = r[3];
    float x2 = r[4], y2 = r[5], x3 = r[6], y3 = r[7];
    float xmin = fminf(fminf(x0, x1), fminf(x2, x3));
    float xmax = fmaxf(fmaxf(x0, x1), fmaxf(x2, x3));
    float ymin = fminf(fminf(y0, y1), fminf(y2, y3));
    float ymax = fmaxf(fmaxf(y0, y1), fmaxf(y2, y3));
    float cx = 0.5f * (xmin + xmax), cy = 0.5f * (ymin + ymax);
    float w = fmaxf(xmax - xmin, 1e-6f), h = fmaxf(ymax - ymin, 1e-6f);
    float gl = 0.5f * (log2f(w * 0.25f) + log2f(h * 0.25f));
    int gli = (int)gl;                                    // trunc toward 0 == astype(int32)
    gli = imin(imax(gli, ws->lvlmin), ws->lvlmax);
    float iw = 1.0f / w, ih = 1.0f / h;
    float C2 = iw * iw, C4 = ih * ih;
    // d^2 = C0 + C1*px + C2*px^2 + C3*py + C4*py^2
    ws->c0[k] = cx * cx * C2 + cy * cy * C4;
    ws->c1[k] = -2.0f * cx * C2;
    ws->c2[k] = C2;
    ws->c3[k] = -2.0f * cy * C4;
    ws->c4[k] = C4;
    ws->glvl[k] = gli;
}

// ---------------- fused distance GEMM (WMMA) + masked argmin ----------------
__global__ void __launch_bounds__(256) k_assign(const float* __restrict__ pts, int N,
                                                Ws* __restrict__ ws) {
    __shared__ float sc0[KGT], sc1[KGT], sc2[KGT], sc3[KGT], sc4[KGT];
    __shared__ int sgl[KGT];
    __shared__ unsigned long long skey[KGT];

    const int tid = threadIdx.x;                          // 256 threads = 8 waves
    sc0[tid] = ws->c0[tid]; sc1[tid] = ws->c1[tid]; sc2[tid] = ws->c2[tid];
    sc3[tid] = ws->c3[tid]; sc4[tid] = ws->c4[tid];
    sgl[tid] = ws->glvl[tid];
    skey[tid] = FULLKEY;
    __syncthreads();

    const int lane = tid & 31;
    const int l4   = lane & 15;
    const int half = lane >> 4;
    const int gwave = blockIdx.x * (blockDim.x >> 5) + (tid >> 5);
    const int nwav  = gridDim.x * (blockDim.x >> 5);
    const int ntiles = (N + 15) >> 4;

    // Hoist loop-invariant B operands (gt coefs) + gt levels into registers.
    v2f  B0r[16];
    v2f  B1r[16];
    int  glr[16];
#pragma unroll
    for (int nt = 0; nt < 16; nt++) {
        const int ng = nt * 16 + l4;
        B0r[nt].x = half ? sc2[ng] : sc0[ng];
        B0r[nt].y = half ? sc3[ng] : sc1[ng];
        B1r[nt].x = half ? 0.0f : sc4[ng];
        B1r[nt].y = 0.0f;
        glr[nt]   = sgl[ng];
    }
    int off8[8];
#pragma unroll
    for (int j = 0; j < 8; j++) off8[j] = j + 8 * half;   // output row offset per VGPR j

    unsigned long long acc[16];
#pragma unroll
    for (int i = 0; i < 16; i++) acc[i] = FULLKEY;

    for (int t = gwave; t < ntiles; t += nwav) {
        const int base = t << 4;
        int pidx = base + l4;
        int ok = pidx < N;
        int pc = ok ? pidx : 0;
        float px = pts[3 * pc + 0];
        float py = pts[3 * pc + 1];
        float st = pts[3 * pc + 2];
        int pl = ok ? ((int)((__float_as_uint(st) >> 23) & 0xFF) - 127) : -1000;

        // prefetch next tile of points (global_prefetch_b8)
        __builtin_prefetch(pts + 3 * (pidx + (nwav << 4)), 0, 0);

        float px2 = px * px, py2 = py * py;
        // A (16 points x 4 feats): lanes 0-15 hold K={0,1}, lanes 16-31 hold K={2,3}
        v2f A0, A1;
        A0.x = half ? px2 : 1.0f;
        A0.y = half ? py  : px;
        A1.x = half ? 0.0f : py2;   // second WMMA covers feature 4 (py^2)
        A1.y = 0.0f;

        // p_lvl and global index of the 8 point-rows this lane owns in D layout
        int plv[8], idx8[8];
#pragma unroll
        for (int j = 0; j < 8; j++) {
            plv[j]  = __shfl(pl, off8[j], 32);
            idx8[j] = base + off8[j];
        }

#pragma unroll
        for (int nt = 0; nt < 16; nt++) {
            v8f c = {0.f, 0.f, 0.f, 0.f, 0.f, 0.f, 0.f, 0.f};
            c = __builtin_amdgcn_wmma_f32_16x16x4_f32(false, A0, false, B0r[nt],
                                                      (short)0, c, false, false);
            c = __builtin_amdgcn_wmma_f32_16x16x4_f32(false, A1, false, B1r[nt],
                                                      (short)0, c, false, false);
            const int gl = glr[nt];
#pragma unroll
            for (int j = 0; j < 8; j++) {
                // Integer clamp of negative d2 (expansion cancellation) to +0:
                // non-negative floats order as signed ints; negatives have sign bit.
                int hb = imax((int)__float_as_uint(c[j]), 0);  // one v_max_i32, no fp ops
                unsigned hi = (plv[j] == gl) ? (unsigned)hb : INVBITS;  // level mask
                unsigned long long key =
                    ((unsigned long long)hi << 32) | (unsigned)idx8[j];
                acc[nt] = key < acc[nt] ? key : acc[nt];  // v_min_u64: min d2, then min idx
            }
        }
    }

    // combine lane halves (points 0-7 vs 8-15) then merge into LDS per-gt table
#pragma unroll
    for (int nt = 0; nt < 16; nt++) {
        unsigned long long v = acc[nt];
        unsigned lo = (unsigned)v, hi = (unsigned)(v >> 32);
        unsigned lo2 = __shfl_xor(lo, 16, 32);
        unsigned hi2 = __shfl_xor(hi, 16, 32);
        unsigned long long o = ((unsigned long long)hi2 << 32) | lo2;
        v = o < v ? o : v;
        if (half == 0) atomicMin(&skey[nt * 16 + l4], v); // ds_min_u64
    }
    __syncthreads();
    atomicMin(&ws->gkeys[tid], skey[tid]);                // one global atomic per gt
}

// ---------------- K^2 conflict resolution + output write ----------------
__global__ void k_final(const int* __restrict__ labels, int N,
                        Ws* __restrict__ ws, float* __restrict__ out) {
    __shared__ unsigned snear[KGT];
    __shared__ unsigned sdb[KGT];
    __shared__ unsigned char sval[KGT], swin[KGT];
    int k = threadIdx.x;
    unsigned long long key = ws->gkeys[k];
    unsigned nearn = (unsigned)(key & 0xFFFFFFFFu);
    unsigned db = (unsigned)(key >> 32);
    int valid = (db != INVBITS);                          // no level-matched point -> invalid
    snear[k] = nearn; sdb[k] = db; sval[k] = (unsigned char)valid;
    __syncthreads();
    unsigned best = INVBITS;
    for (int j = 0; j < KGT; j++)
        if (sval[j] && snear[j] == nearn) best = (sdb[j] < best) ? sdb[j] : best;
    swin[k] = (unsigned char)(valid && (db == best));
    __syncthreads();
    int ow = KGT;
    for (int j = 0; j < KGT; j++)
        if (sval[j] && snear[j] == nearn && swin[j]) { ow = j; break; }
    if (valid) {                                          // winner always exists here
        out[nearn]     = (float)(ow + 1);
        out[N + nearn] = (float)labels[ow];
    }
}

extern "C" void kernel_launch(void* const* d_in, const int* in_sizes, int n_in,
                              void* d_out, int out_size, void* d_ws, size_t ws_size,
                              hipStream_t stream) {
    const float* pts = (const float*)d_in[0];   // (N,3)
    const float* gtb = (const float*)d_in[1];   // (K,8)
    const int* labels = (const int*)d_in[2];    // (K,)
    int N = in_sizes[0] / 3;
    float* out = (float*)d_out;                 // [assigned_gt_inds | assigned_labels]
    Ws* ws = (Ws*)d_ws;
    int twoN = 2 * N;

    k_init<<<(twoN + 255) / 256, 256, 0, stream>>>(out, twoN, ws);
    k_lvl<<<256, 256, 0, stream>>>(pts, N, ws);
    k_prep<<<1, 256, 0, stream>>>(gtb, ws);
    k_assign<<<256, 256, 0, stream>>>(pts, N, ws);
    k_final<<<1, 256, 0, stream>>>(labels, N, ws, out);
}